// NCC_52862457480005
// MI455X (gfx1250) — compile-verified
//
#include <hip/hip_runtime.h>

typedef float v2f __attribute__((ext_vector_type(2)));
typedef float v8f __attribute__((ext_vector_type(8)));
typedef unsigned int u32x4 __attribute__((ext_vector_type(4)));
typedef int i32x4 __attribute__((ext_vector_type(4)));
typedef int i32x8 __attribute__((ext_vector_type(8)));

#define NCC_EPS 1.1920928955078125e-7f

#if __has_builtin(__builtin_amdgcn_tensor_load_to_lds)
#define NCC_HAS_TDM 1
#endif

__device__ __forceinline__ void ncc_wait_tensor0() {
#if __has_builtin(__builtin_amdgcn_s_wait_tensorcnt)
    __builtin_amdgcn_s_wait_tensorcnt(0);
#else
    asm volatile("s_wait_tensorcnt 0x0" ::: "memory");
#endif
}

#ifdef NCC_HAS_TDM
// Issue one TDM load of a 2D tile: 16 rows x W cols of f32, row stride N
// elements, from gsrc (tile origin) into LDS at byte offset lds_off (rows
// packed at stride W).  Descriptor per CDNA5 ISA ch.8 (D# groups 0/1).
__device__ __forceinline__ void ncc_tdm_load_tile(const float* gsrc,
                                                  unsigned lds_off,
                                                  int W, int N) {
    unsigned long long ga = (unsigned long long)(uintptr_t)gsrc;
    u32x4 g0;
    g0.x = 1u;                                            // count=1, user mode
    g0.y = lds_off;                                       // lds_addr (bytes)
    g0.z = (unsigned)(ga & 0xFFFFFFFFu);                  // global_addr[31:0]
    g0.w = (unsigned)((ga >> 32) & 0x01FFFFFFu) | (2u << 30); // [56:32] | type=2

    unsigned td0 = (unsigned)N;                           // tensor_dim0
    unsigned td1 = (unsigned)(N * N);                     // tensor_dim1
    i32x8 g1;
    g1[0] = (int)(2u << 16);                              // data_size=4B, mask=0
    g1[1] = (int)((td0 & 0xFFFFu) << 16);                 // dim0[15:0] @ [63:48]
    g1[2] = (int)((td0 >> 16) | ((td1 & 0xFFFFu) << 16)); // dim0[31:16]|dim1[15:0]
    g1[3] = (int)((td1 >> 16) | ((unsigned)W << 16));     // dim1[31:16]|tile_dim0
    g1[4] = 16;                                           // tile_dim1=16, tile_dim2=0
    g1[5] = N;                                            // dim0_stride[31:0]
    g1[6] = 0;                                            // stride hi | dim1_stride lo
    g1[7] = 0;

    i32x4 z4 = {0, 0, 0, 0};
#if __has_include(<hip/amd_detail/amd_gfx1250_TDM.h>)
    i32x8 z8 = {0, 0, 0, 0, 0, 0, 0, 0};
    __builtin_amdgcn_tensor_load_to_lds(g0, g1, z4, z4, z8, 0);
#else
    __builtin_amdgcn_tensor_load_to_lds(g0, g1, z4, z4, 0);
#endif
}
#endif // NCC_HAS_TDM

// ---------------------------------------------------------------------------
// Pass 1: box filter along X for one moment channel via V_WMMA_F32_16X16X4_F32.
// One wave per 16x16 tile (16 flattened (z,y) lines x 16 x-outputs):
//   out(16x16) = sum_{c} A_c(16x4 inputs) * B_c(4x16 banded coeffs)
// CH: 0=I, 1=J, 2=I*I, 3=J*J, 4=I*J (products fused into the A-operand build).
// Interior tiles are staged global->LDS by the Tensor Data Mover.
// ---------------------------------------------------------------------------
template<int P, int CH>
__global__ __launch_bounds__(32)
void ncc_pass_x_wmma(const float* __restrict__ img1, const float* __restrict__ img2,
                     float* __restrict__ outA, int N) {
    constexpr int W  = 16 + 2 * P;             // tile width (24/22/20)
    constexpr int NC = (W + 3) / 4;            // K=4 chunks
    constexpr bool NI = (CH == 0 || CH == 2 || CH == 4);
    constexpr bool NJ = (CH == 1 || CH == 3 || CH == 4);

    __shared__ float tI[16 * 24 + 16];         // rows packed at stride W + zero tail
    __shared__ float tJ[16 * 24 + 16];

    const int lane    = threadIdx.x;
    const int xbase   = blockIdx.x * 16;
    const int rowbase = blockIdx.y * 16;       // flattened (z*N + y)

    // Zero the tail so the last row's K-chunk over-read hits 0, not garbage.
    if (lane < 16) {
        if (NI) tI[16 * W + lane] = 0.0f;
        if (NJ) tJ[16 * W + lane] = 0.0f;
    }

    const bool interior = (xbase - P >= 0) && (xbase + 16 + P <= N);
#ifdef NCC_HAS_TDM
    if (interior) {
        size_t off = (size_t)rowbase * (size_t)N + (size_t)(xbase - P);
        if (NI) ncc_tdm_load_tile(img1 + off, (unsigned)(uintptr_t)&tI[0], W, N);
        if (NJ) ncc_tdm_load_tile(img2 + off, (unsigned)(uintptr_t)&tJ[0], W, N);
        ncc_wait_tensor0();
    } else
#else
    (void)interior;
#endif
    {
        // Manual zero-padded fill (16*W is a multiple of 32 for W=24/22/20).
        for (int idx = lane; idx < 16 * W; idx += 32) {
            int r = idx / W, c = idx - r * W;
            int gx = xbase - P + c;
            bool ok = (gx >= 0 && gx < N);
            size_t off = (size_t)(rowbase + r) * (size_t)N + (size_t)(ok ? gx : 0);
            if (NI) tI[idx] = ok ? img1[off] : 0.0f;
            if (NJ) tJ[idx] = ok ? img2[off] : 0.0f;
        }
    }
    __syncthreads();

    const float invk = 1.0f / (float)(2 * P + 1);
    const int hi  = lane >> 4;                 // lane half
    const int l15 = lane & 15;

    v8f acc = {0.f, 0.f, 0.f, 0.f, 0.f, 0.f, 0.f, 0.f};

#pragma unroll
    for (int c = 0; c < NC; ++c) {
        // A 16x4 f32 layout: VGPR0 = K={0|2}, VGPR1 = K={1|3} per lane half.
        int t0 = c * 4 + (hi << 1);
        int t1 = t0 + 1;
        float a0, a1;
        if constexpr (CH == 0) {
            a0 = tI[l15 * W + t0];             a1 = tI[l15 * W + t1];
        } else if constexpr (CH == 1) {
            a0 = tJ[l15 * W + t0];             a1 = tJ[l15 * W + t1];
        } else if constexpr (CH == 2) {
            float i0 = tI[l15 * W + t0], i1 = tI[l15 * W + t1];
            a0 = i0 * i0;                      a1 = i1 * i1;
        } else if constexpr (CH == 3) {
            float j0 = tJ[l15 * W + t0], j1 = tJ[l15 * W + t1];
            a0 = j0 * j0;                      a1 = j1 * j1;
        } else {
            float i0 = tI[l15 * W + t0], i1 = tI[l15 * W + t1];
            float j0 = tJ[l15 * W + t0], j1 = tJ[l15 * W + t1];
            a0 = i0 * j0;                      a1 = i1 * j1;
        }
        // B 4x16 banded coefficients: w[t][n] = invk iff n <= t <= n + 2P.
        float b0 = (t0 >= l15 && t0 <= l15 + 2 * P) ? invk : 0.0f;
        float b1 = (t1 >= l15 && t1 <= l15 + 2 * P) ? invk : 0.0f;
        v2f a = {a0, a1};
        v2f b = {b0, b1};
        acc = __builtin_amdgcn_wmma_f32_16x16x4_f32(
            false, a, false, b, (short)0, acc, false, false);
    }

    // D layout: VGPR g -> M = g (lanes 0-15) / g+8 (lanes 16-31), N = lane&15.
#pragma unroll
    for (int g = 0; g < 8; ++g) {
        int m = g + (hi ? 8 : 0);
        outA[(size_t)(rowbase + m) * (size_t)N + (size_t)(xbase + l15)] = acc[g];
    }
}

// ---------------------------------------------------------------------------
// Pass 2: box filter along Y; thread owns one (z,x) line, 16 outputs,
// register-blocked sliding window, x fastest -> coalesced.
// ---------------------------------------------------------------------------
template<int P>
__global__ __launch_bounds__(256)
void ncc_pass_y(const float* __restrict__ in, float* __restrict__ out, int N) {
    constexpr int W = 16 + 2 * P;
    int g = blockIdx.x * 256 + threadIdx.x;   // over N*N (z,x)
    int x = g % N, z = g / N;
    int y0 = blockIdx.y * 16;
    const float invk = 1.0f / (float)(2 * P + 1);

    float v[W];
#pragma unroll
    for (int i = 0; i < W; ++i) {
        int y = y0 - P + i;
        v[i] = (y >= 0 && y < N) ? in[((size_t)z * N + y) * (size_t)N + x] : 0.0f;
    }
#pragma unroll
    for (int o = 0; o < 16; ++o) {
        float s = v[o];
#pragma unroll
        for (int t = 1; t < 2 * P + 1; ++t) s += v[o + t];
        out[((size_t)z * N + (y0 + o)) * (size_t)N + x] = s * invk;
    }
}

// ---------------------------------------------------------------------------
// Pass 3: box filter along Z for all 5 channels + cc + block reduction.
// ---------------------------------------------------------------------------
template<int P>
__global__ __launch_bounds__(256)
void ncc_pass_z_cc(const float* __restrict__ B5, size_t V, int N,
                   float* __restrict__ acc) {
    constexpr int W = 16 + 2 * P;
    int g = blockIdx.x * 256 + threadIdx.x;   // over N*N (y,x)
    int x = g % N, y = g / N;
    int z0 = blockIdx.y * 16;
    const float invk = 1.0f / (float)(2 * P + 1);

    float o5[5][16];
#pragma unroll
    for (int ch = 0; ch < 5; ++ch) {
        const float* src = B5 + (size_t)ch * V;
        float v[W];
#pragma unroll
        for (int i = 0; i < W; ++i) {
            int z = z0 - P + i;
            v[i] = (z >= 0 && z < N) ? src[((size_t)z * N + y) * (size_t)N + x] : 0.0f;
        }
#pragma unroll
        for (int o = 0; o < 16; ++o) {
            float s = v[o];
#pragma unroll
            for (int t = 1; t < 2 * P + 1; ++t) s += v[o + t];
            o5[ch][o] = s * invk;
        }
    }

    float lsum = 0.0f;
#pragma unroll
    for (int o = 0; o < 16; ++o) {
        float mu1 = o5[0][o], mu2 = o5[1][o];
        float s1  = o5[2][o] - mu1 * mu1;
        float s2  = o5[3][o] - mu2 * mu2;
        float s12 = o5[4][o] - mu1 * mu2;
        float cc  = (s12 * s12) / fmaxf(s1 * s2, NCC_EPS);
        lsum += cc;
    }

    __shared__ float red[256];
    int tid = threadIdx.x;
    red[tid] = lsum;
    __syncthreads();
#pragma unroll
    for (int s = 128; s > 0; s >>= 1) {
        if (tid < s) red[tid] += red[tid + s];
        __syncthreads();
    }
    if (tid == 0) atomicAdd(acc, red[0]);
}

// ---------------------------------------------------------------------------
// avg_pool3d k=3 s=2 p=1, count_include_pad=False.
// ---------------------------------------------------------------------------
__global__ __launch_bounds__(256)
void ncc_pool3(const float* __restrict__ in, float* __restrict__ out, int N) {
    int M = N >> 1;
    size_t tot = (size_t)M * M * M;
    size_t o = (size_t)blockIdx.x * 256 + threadIdx.x;
    if (o >= tot) return;
    int ox = (int)(o % M);
    int t  = (int)(o / M);
    int oy = t % M;
    int oz = t / M;
    float s = 0.0f;
    int c = 0;
    for (int dz = -1; dz <= 1; ++dz) {
        int z = 2 * oz + dz;
        if ((unsigned)z >= (unsigned)N) continue;
        for (int dy = -1; dy <= 1; ++dy) {
            int y = 2 * oy + dy;
            if ((unsigned)y >= (unsigned)N) continue;
            size_t base = ((size_t)z * N + y) * (size_t)N;
            for (int dx = -1; dx <= 1; ++dx) {
                int x = 2 * ox + dx;
                if ((unsigned)x >= (unsigned)N) continue;
                s += in[base + x];
                ++c;
            }
        }
    }
    out[o] = s / (float)c;
}

__global__ void ncc_zero4(float* a) {
    if (threadIdx.x < 4) a[threadIdx.x] = 0.0f;
}

__global__ void ncc_finish(const float* __restrict__ acc, float* __restrict__ out) {
    float m0 = acc[0] * (1.0f / 7077888.0f);
    float m1 = acc[1] * (1.0f / 884736.0f);
    float m2 = acc[2] * (1.0f / 110592.0f);
    out[0] = -(m0 + m1 + m2) * (1.0f / 3.0f);
}

// ---------------------------------------------------------------------------
// Host-side orchestration.
// ---------------------------------------------------------------------------
template<int P, int CH>
static inline void ncc_run_ch(const float* i1, const float* i2, float* A, float* B,
                              int N, size_t V, hipStream_t stream) {
    dim3 g1(N / 16, (N * N) / 16);
    dim3 g2((N * N) / 256, N / 16);
    ncc_pass_x_wmma<P, CH><<<g1, 32, 0, stream>>>(i1, i2, A, N);
    ncc_pass_y<P><<<g2, 256, 0, stream>>>(A, B + (size_t)CH * V, N);
}

template<int P>
static void ncc_run_scale(const float* img1, const float* img2, float* A, float* B,
                          float* acc_s, int N, hipStream_t stream) {
    size_t V = (size_t)N * N * N;
    ncc_run_ch<P, 0>(img1, img2, A, B, N, V, stream);
    ncc_run_ch<P, 1>(img1, img2, A, B, N, V, stream);
    ncc_run_ch<P, 2>(img1, img2, A, B, N, V, stream);
    ncc_run_ch<P, 3>(img1, img2, A, B, N, V, stream);
    ncc_run_ch<P, 4>(img1, img2, A, B, N, V, stream);
    dim3 g2((N * N) / 256, N / 16);
    ncc_pass_z_cc<P><<<g2, 256, 0, stream>>>(B, V, N, acc_s);
}

extern "C" void kernel_launch(void* const* d_in, const int* in_sizes, int n_in,
                              void* d_out, int out_size, void* d_ws, size_t ws_size,
                              hipStream_t stream) {
    (void)in_sizes; (void)n_in; (void)out_size; (void)ws_size;

    const float* pred   = (const float*)d_in[0];
    const float* target = (const float*)d_in[1];
    float* w = (float*)d_ws;

    const size_t V0 = 192ull * 192 * 192;   // 7,077,888
    const size_t V1 = 96ull * 96 * 96;      //   884,736
    const size_t V2 = 48ull * 48 * 48;      //   110,592

    // Workspace layout (floats): B5 | A | pooled s1 (x2) | pooled s2 (x2) | acc
    float* B   = w;               // 5*V0
    float* A   = w + 5 * V0;      // V0
    float* p1a = w + 6 * V0;      // V1
    float* p1b = p1a + V1;        // V1
    float* p2a = p1b + V1;        // V2
    float* p2b = p2a + V2;        // V2
    float* acc = p2b + V2;        // 4

    ncc_zero4<<<1, 32, 0, stream>>>(acc);

    // Scale 0: 192^3, win 9 (P=4)
    ncc_run_scale<4>(pred, target, A, B, acc + 0, 192, stream);
    ncc_pool3<<<(unsigned)(V1 / 256), 256, 0, stream>>>(pred,   p1a, 192);
    ncc_pool3<<<(unsigned)(V1 / 256), 256, 0, stream>>>(target, p1b, 192);

    // Scale 1: 96^3, win 7 (P=3)
    ncc_run_scale<3>(p1a, p1b, A, B, acc + 1, 96, stream);
    ncc_pool3<<<(unsigned)(V2 / 256), 256, 0, stream>>>(p1a, p2a, 96);
    ncc_pool3<<<(unsigned)(V2 / 256), 256, 0, stream>>>(p1b, p2b, 96);

    // Scale 2: 48^3, win 5 (P=2)
    ncc_run_scale<2>(p2a, p2b, A, B, acc + 2, 48, stream);

    ncc_finish<<<1, 1, 0, stream>>>(acc, (float*)d_out);
}